// DeepSeekDecodeAttention_10101763080531
// MI455X (gfx1250) — compile-verified
//
#include <hip/hip_runtime.h>
#include <hip/hip_bf16.h>

typedef __attribute__((ext_vector_type(2))) float v2f;
typedef __attribute__((ext_vector_type(4))) float v4f;
typedef __attribute__((ext_vector_type(8))) float v8f;
typedef __attribute__((ext_vector_type(4))) unsigned int u32x4;
typedef __attribute__((ext_vector_type(4))) int i32x4;
typedef __attribute__((ext_vector_type(8))) int i32x8;

#define HIDDEN   7168
#define N_HEADS  128
#define HEAD_DIM 128
#define Q_LORA   1536
#define KV_LORA  512
#define CACHE_LEN 512
#define BATCH    8

#ifdef __has_builtin
# if __has_builtin(__builtin_amdgcn_tensor_load_to_lds)
#  define HAVE_TDM 1
# endif
#endif
#ifndef HAVE_TDM
# define HAVE_TDM 0
#endif

// ---------------------------------------------------------------- utilities
__global__ void zero_f32(float* __restrict__ p, int n) {
  int i = blockIdx.x * blockDim.x + threadIdx.x;
  int stride = gridDim.x * blockDim.x;
  for (; i < n; i += stride) p[i] = 0.0f;
}

__device__ __forceinline__ void atomAddF(float* p, float v) {
  unsafeAtomicAdd(p, v);  // hardware global_atomic_add_f32
}

#if HAVE_TDM
// Issue a TDM 2D tile load: tile (tile_w x tile_h) of 4-byte elements from a
// row-major matrix with row stride `row_stride` elements, into LDS at lds_addr.
// D# packing per CDNA5 ISA §8.3 (group0) / §8.4 (group1).
__device__ __forceinline__ void tdm_load_tile_f32(unsigned int lds_addr,
                                                  const float* gptr,
                                                  unsigned int tile_w,
                                                  unsigned int tile_h,
                                                  unsigned long long row_stride) {
  unsigned long long ga = (unsigned long long)(uintptr_t)gptr;
  u32x4 g0;
  g0[0] = 1u;                                   // count=1, user D#, no gather
  g0[1] = lds_addr;                             // lds_addr [63:32]
  g0[2] = (unsigned int)ga;                     // global_addr[31:0]
  g0[3] = (unsigned int)((ga >> 32) & 0x01FFFFFFu) | (2u << 30);  // addr[56:32], type=2
  unsigned int td0 = (unsigned int)row_stride;  // tensor_dim0 >= tile_w
  unsigned int td1 = tile_h;                    // tensor_dim1 (tile always in-bounds)
  i32x8 g1;
  g1[0] = (int)(2u << 16);                      // wg_mask=0, data_size=2 (4B), no flags
  g1[1] = (int)((td0 & 0xFFFFu) << 16);         // barrier_addr=0 | tensor_dim0[15:0]
  g1[2] = (int)((td0 >> 16) | ((td1 & 0xFFFFu) << 16));  // dim0[31:16] | dim1[15:0]
  g1[3] = (int)((td1 >> 16) | (tile_w << 16));  // dim1[31:16] | tile_dim0
  g1[4] = (int)(tile_h & 0xFFFFu);              // tile_dim1 | tile_dim2=0
  g1[5] = (int)(unsigned int)row_stride;        // tensor_dim0_stride[31:0]
  g1[6] = (int)(unsigned int)(row_stride >> 32);// stride[47:32] | dim1_stride[15:0]=0
  g1[7] = 0;
  i32x4 z4 = {0, 0, 0, 0};
#if __clang_major__ >= 23
  i32x8 z8 = {0, 0, 0, 0, 0, 0, 0, 0};
  __builtin_amdgcn_tensor_load_to_lds(g0, g1, z4, z4, z8, 0);
#else
  __builtin_amdgcn_tensor_load_to_lds(g0, g1, z4, z4, 0);
#endif
}
#endif  // HAVE_TDM

// ---------------------------------------------------------------- WMMA GEMM
// C(8 x N) += A(8 x K) @ B(K x N), all f32 row-major.
// One wave per 16-wide N tile; M padded 8->16 with zero rows.
// Split-K across blockIdx.y, partials combined with f32 atomics.
template <bool NT>
__global__ void gemm8_wmma(const float* __restrict__ A,
                           const float* __restrict__ B,
                           float* __restrict__ C,
                           int K, int N, int k_chunk) {
  const int n0   = blockIdx.x * 16;
  const int kbeg = blockIdx.y * k_chunk;
  int kend = kbeg + k_chunk;
  if (kend > K) kend = K;

  const int  lane = threadIdx.x;       // 0..31
  const int  m    = lane & 15;
  const bool hi   = lane >= 16;
  const float mmask = (m < BATCH) ? 1.0f : 0.0f;   // zero-pad rows 8..15
  const float* Ap = A + (size_t)(m & (BATCH - 1)) * (size_t)K;
  const float* Bp = B + (size_t)n0 + (size_t)m + (size_t)(hi ? 2 : 0) * (size_t)N;

  v8f acc = {0.f, 0.f, 0.f, 0.f, 0.f, 0.f, 0.f, 0.f};

#pragma unroll 4
  for (int k = kbeg; k < kend; k += 4) {
    v4f av = *(const v4f*)(Ap + k);
    float a0 = (hi ? av.z : av.x) * mmask;
    float a1 = (hi ? av.w : av.y) * mmask;

    const float* bp = Bp + (size_t)k * (size_t)N;
    float b0, b1;
    if (NT) {
      b0 = __builtin_nontemporal_load(bp);
      b1 = __builtin_nontemporal_load(bp + N);
    } else {
      b0 = bp[0];
      b1 = bp[N];
    }
    if (k + 64 < kend) __builtin_prefetch(bp + (size_t)64 * (size_t)N, 0, 3);

    v2f a = {a0, a1};
    v2f b = {b0, b1};
    acc = __builtin_amdgcn_wmma_f32_16x16x4_f32(false, a, false, b,
                                                (short)0, acc, false, false);
  }

#pragma unroll
  for (int r = 0; r < 8; ++r) {
    int mm = hi ? (r + 8) : r;
    if (mm < BATCH) atomAddF(&C[(size_t)mm * (size_t)N + n0 + m], acc[r]);
  }
}

// ------------------------------------------------- WMMA GEMM with TDM staging
// Same math as gemm8_wmma but B tiles (64 rows x 16 cols, 4 KB) are streamed
// into LDS by the Tensor Data Mover, double-buffered, overlapped with WMMA.
// k_chunk must be a multiple of 64. One wave per block (no barriers needed).
__global__ void gemm8_wmma_tdm(const float* __restrict__ A,
                               const float* __restrict__ B,
                               float* __restrict__ C,
                               int K, int N, int k_chunk) {
  __shared__ float bs[2][64 * 16];
  const int n0   = blockIdx.x * 16;
  const int kbeg = blockIdx.y * k_chunk;
  int kend = kbeg + k_chunk;
  if (kend > K) kend = K;

  const int  lane = threadIdx.x;
  const int  m    = lane & 15;
  const bool hi   = lane >= 16;
  const int  xoff = hi ? 2 : 0;
  const float mmask = (m < BATCH) ? 1.0f : 0.0f;
  const float* Ap = A + (size_t)(m & (BATCH - 1)) * (size_t)K;

  v8f acc = {0.f, 0.f, 0.f, 0.f, 0.f, 0.f, 0.f, 0.f};

#if HAVE_TDM && defined(__HIP_DEVICE_COMPILE__)
  const unsigned int lds0 = (unsigned int)(uintptr_t)(&bs[0][0]);
  const unsigned int lds1 = (unsigned int)(uintptr_t)(&bs[1][0]);

  // prime buffer 0
  tdm_load_tile_f32(lds0, B + (size_t)kbeg * (size_t)N + n0, 16u, 64u,
                    (unsigned long long)N);

  for (int chunk = kbeg; chunk < kend; chunk += 64) {
    const int cur = ((chunk - kbeg) >> 6) & 1;
    const bool has_next = (chunk + 64) < kend;
    if (has_next) {
      tdm_load_tile_f32(cur ? lds0 : lds1,
                        B + (size_t)(chunk + 64) * (size_t)N + n0, 16u, 64u,
                        (unsigned long long)N);
      __builtin_amdgcn_s_wait_tensorcnt(1);  // current tile resident
    } else {
      __builtin_amdgcn_s_wait_tensorcnt(0);
    }
    const float* bsc = bs[cur];
#pragma unroll
    for (int kk = 0; kk < 64; kk += 4) {
      v4f av = *(const v4f*)(Ap + chunk + kk);
      float a0 = (hi ? av.z : av.x) * mmask;
      float a1 = (hi ? av.w : av.y) * mmask;
      float b0 = bsc[(kk + xoff) * 16 + m];
      float b1 = bsc[(kk + xoff + 1) * 16 + m];
      v2f a = {a0, a1};
      v2f b = {b0, b1};
      acc = __builtin_amdgcn_wmma_f32_16x16x4_f32(false, a, false, b,
                                                  (short)0, acc, false, false);
    }
  }
#else
  // Fallback (also used for host-pass semantic checking): plain global loads.
  const float* Bp = B + (size_t)n0 + (size_t)m + (size_t)xoff * (size_t)N;
  for (int k = kbeg; k < kend; k += 4) {
    v4f av = *(const v4f*)(Ap + k);
    float a0 = (hi ? av.z : av.x) * mmask;
    float a1 = (hi ? av.w : av.y) * mmask;
    const float* bp = Bp + (size_t)k * (size_t)N;
    v2f a = {a0, a1};
    v2f b = {bp[0], bp[N]};
    acc = __builtin_amdgcn_wmma_f32_16x16x4_f32(false, a, false, b,
                                                (short)0, acc, false, false);
  }
#endif

#pragma unroll
  for (int r = 0; r < 8; ++r) {
    int mm = hi ? (r + 8) : r;
    if (mm < BATCH) atomAddF(&C[(size_t)mm * (size_t)N + n0 + m], acc[r]);
  }
}

// ---------------------------------------------------------------- attention
// One block per (head, batch): scores = (q . K_cache)*scale over 512 keys,
// softmax, weighted V sum, accumulated into attn[] (which already holds v_new,
// because softmax over the length-1 new-token axis is identically 1).
__global__ void mla_attention(const float* __restrict__ q,     // (B, H*D)
                              const float* __restrict__ Kc,    // (B,H,S,D)
                              const float* __restrict__ Vc,    // (B,H,S,D)
                              float* __restrict__ attn) {      // (B, H*D)
  const int h = blockIdx.x;
  const int b = blockIdx.y;
  const int tid  = threadIdx.x;          // 256 threads = 8 waves
  const int wave = tid >> 5;
  const int lane = tid & 31;

  __shared__ float sq[HEAD_DIM];
  __shared__ float sc[CACHE_LEN];
  __shared__ float red[8];
  __shared__ float partial[HEAD_DIM];

  const size_t bh = (size_t)b * N_HEADS + h;
  const float* qrow = q + (size_t)b * (N_HEADS * HEAD_DIM) + (size_t)h * HEAD_DIM;
  const float* Kb = Kc + bh * CACHE_LEN * HEAD_DIM;
  const float* Vb = Vc + bh * CACHE_LEN * HEAD_DIM;

  if (tid < HEAD_DIM) sq[tid] = qrow[tid];
  __syncthreads();

  const float q0 = sq[lane * 4 + 0], q1 = sq[lane * 4 + 1];
  const float q2 = sq[lane * 4 + 2], q3 = sq[lane * 4 + 3];
  const float scale = 0.088388347648318447f;  // 1/sqrt(128)

  // --- scores: each wave streams whole 512B key rows (fully coalesced) ---
  for (int key = wave; key < CACHE_LEN; key += 8) {
    v4f kv = __builtin_nontemporal_load(
        (const v4f*)(Kb + (size_t)key * HEAD_DIM + lane * 4));
    float d = kv.x * q0 + kv.y * q1 + kv.z * q2 + kv.w * q3;
#pragma unroll
    for (int off = 16; off; off >>= 1) d += __shfl_xor(d, off, 32);
    if (lane == 0) sc[key] = d * scale;
  }
  __syncthreads();

  // --- softmax over 512 scores in LDS ---
  float mx = -INFINITY;
  for (int i = tid; i < CACHE_LEN; i += 256) mx = fmaxf(mx, sc[i]);
#pragma unroll
  for (int off = 16; off; off >>= 1) mx = fmaxf(mx, __shfl_xor(mx, off, 32));
  if (lane == 0) red[wave] = mx;
  __syncthreads();
  mx = red[0];
#pragma unroll
  for (int w = 1; w < 8; ++w) mx = fmaxf(mx, red[w]);
  __syncthreads();

  float s = 0.0f;
  for (int i = tid; i < CACHE_LEN; i += 256) {
    float e = __expf(sc[i] - mx);
    sc[i] = e;
    s += e;
  }
#pragma unroll
  for (int off = 16; off; off >>= 1) s += __shfl_xor(s, off, 32);
  if (lane == 0) red[wave] = s;
  __syncthreads();
  s = red[0] + red[1] + red[2] + red[3] + red[4] + red[5] + red[6] + red[7];
  const float inv = 1.0f / s;

  // --- weighted V: 2-way key split, coalesced across d ---
  const int d = tid & (HEAD_DIM - 1);
  const int kstart = (tid < HEAD_DIM) ? 0 : (CACHE_LEN / 2);
  float acc = 0.0f;
  for (int key = kstart; key < kstart + CACHE_LEN / 2; ++key)
    acc += sc[key] *
           __builtin_nontemporal_load(Vb + (size_t)key * HEAD_DIM + d);
  if (tid >= HEAD_DIM) partial[d] = acc;
  __syncthreads();
  if (tid < HEAD_DIM) {
    float tot = (acc + partial[d]) * inv;
    attn[(size_t)b * (N_HEADS * HEAD_DIM) + (size_t)h * HEAD_DIM + d] += tot;
  }
}

// ---------------------------------------------------------------- launch
extern "C" void kernel_launch(void* const* d_in, const int* in_sizes, int n_in,
                              void* d_out, int out_size, void* d_ws, size_t ws_size,
                              hipStream_t stream) {
  const float* x        = (const float*)d_in[0];
  const float* k_cache  = (const float*)d_in[1];
  const float* v_cache  = (const float*)d_in[2];
  const float* Wq_down  = (const float*)d_in[3];
  const float* Wq_up    = (const float*)d_in[4];
  const float* Wkv_down = (const float*)d_in[5];
  // d_in[6] = Wk_up: unused — softmax over the length-1 new-token axis == 1,
  // so new_out == v_new and k_new never affects the output (saves 33.6 MB).
  const float* Wv_up    = (const float*)d_in[7];
  const float* Wo       = (const float*)d_in[8];
  float* out = (float*)d_out;

  float* q_lat = (float*)d_ws;                         // 8 x 1536
  float* c_kv  = q_lat + BATCH * Q_LORA;               // 8 x 512
  float* qf    = c_kv + BATCH * KV_LORA;               // 8 x 16384
  float* attn  = qf + BATCH * N_HEADS * HEAD_DIM;      // 8 x 16384 (v_new + cache_out)
  const int ws_elems = BATCH * (Q_LORA + KV_LORA + 2 * N_HEADS * HEAD_DIM);

  zero_f32<<<256, 256, 0, stream>>>((float*)d_ws, ws_elems);
  zero_f32<<<64, 256, 0, stream>>>(out, out_size);

  // q_lat = x @ Wq_down        (K=7168, N=1536, split-K=4)
  gemm8_wmma<false><<<dim3(Q_LORA / 16, 4), 32, 0, stream>>>(
      x, Wq_down, q_lat, HIDDEN, Q_LORA, HIDDEN / 4);
  // c_kv = x @ Wkv_down        (K=7168, N=512, split-K=4)
  gemm8_wmma<false><<<dim3(KV_LORA / 16, 4), 32, 0, stream>>>(
      x, Wkv_down, c_kv, HIDDEN, KV_LORA, HIDDEN / 4);
  // q = q_lat @ Wq_up          (K=1536, N=16384, split-K=2, TDM-staged B)
  gemm8_wmma_tdm<<<dim3(N_HEADS * HEAD_DIM / 16, 2), 32, 0, stream>>>(
      q_lat, Wq_up, qf, Q_LORA, N_HEADS * HEAD_DIM, Q_LORA / 2);
  // attn = v_new = c_kv @ Wv_up  (K=512, N=16384)
  gemm8_wmma<false><<<dim3(N_HEADS * HEAD_DIM / 16, 1), 32, 0, stream>>>(
      c_kv, Wv_up, attn, KV_LORA, N_HEADS * HEAD_DIM, KV_LORA);
  // attn += softmax(q.K)·V per (b,h)
  mla_attention<<<dim3(N_HEADS, BATCH), 256, 0, stream>>>(qf, k_cache, v_cache, attn);
  // out = attn @ Wo            (K=16384, N=7168, split-K=8, TDM-staged B)
  gemm8_wmma_tdm<<<dim3(HIDDEN / 16, 8), 32, 0, stream>>>(
      attn, Wo, out, N_HEADS * HEAD_DIM, HIDDEN, N_HEADS * HEAD_DIM / 8);
}